// S4DKernel_63513976374074
// MI455X (gfx1250) — compile-verified
//
#include <hip/hip_runtime.h>

#define H_DIM 1024
#define N_DIM 64
#define L_DIM 2048

typedef float v2f __attribute__((ext_vector_type(2)));
typedef float v8f __attribute__((ext_vector_type(8)));

__device__ __forceinline__ void cmul(float& xr, float& xi, float yr, float yi) {
    float tr = xr * yr - xi * yi;
    xi = xr * yi + xi * yr;
    xr = tr;
}

// One wave32 per head h. Computes out[h, 0:2048] via two real f32 WMMA matmuls:
//   out[l0 + 16*l1] = (Qr·Pr + Qi·(-Pi))[l0, l1],  Q:16x64, P:64x128 (two 64-col passes)
__global__ __launch_bounds__(32) void s4d_wmma_kernel(
    const float* __restrict__ log_dt,      // (H)
    const float* __restrict__ C_real,      // (H, N, 2)
    const float* __restrict__ log_A_real,  // (H, N)
    const float* __restrict__ A_imag,      // (H, N)
    float* __restrict__ out)               // (H, L)
{
    // Padded LDS staging (stride 65 floats -> conflict-free column access)
    __shared__ float sQr[16 * 65];
    __shared__ float sQi[16 * 65];
    __shared__ float sPr[64 * 65];
    __shared__ float sPi[64 * 65];   // holds -Im(P): minus folded in here

    const int h       = blockIdx.x;
    const int lane    = threadIdx.x;   // 0..31 (wave32)
    const int halfSel = lane >> 4;     // selects K-pair / M-half
    const int mn      = lane & 15;

    const float dt = expf(log_dt[h]);

    // Per-lane setup for its two n-rows: n = lane, lane+32
    float wr[2], wi[2], W16r[2], W16i[2], pr[2], pi[2];
    #pragma unroll
    for (int s = 0; s < 2; ++s) {
        const int n   = lane + 32 * s;
        const float lar = log_A_real[h * N_DIM + n];
        const float aim = A_imag[h * N_DIM + n];
        const float cr  = C_real[(h * N_DIM + n) * 2 + 0];
        const float ci  = C_real[(h * N_DIM + n) * 2 + 1];

        const float ar  = -expf(lar);          // Re(A)
        const float dAr = ar * dt, dAi = aim * dt;  // dtA

        const float er = expf(dAr);            // w = exp(dtA)
        wr[s] = er * cosf(dAi);
        wi[s] = er * sinf(dAi);

        // C2 = C * (exp(dtA)-1) / A ;  /A = conj(A)/|A|^2 ; fold in the final 2x
        const float e1r = wr[s] - 1.0f, e1i = wi[s];
        const float inv = 1.0f / (ar * ar + aim * aim);
        const float tr  = (e1r * ar + e1i * aim) * inv;
        const float ti  = (e1i * ar - e1r * aim) * inv;
        pr[s] = 2.0f * (cr * tr - ci * ti);    // running P state starts at 2*C2
        pi[s] = 2.0f * (cr * ti + ci * tr);

        const float er16 = expf(16.0f * dAr);  // W16 = exp(16*dtA) (exact, not w^16)
        W16r[s] = er16 * cosf(16.0f * dAi);
        W16i[s] = er16 * sinf(16.0f * dAi);
    }

    // Build Q[l0, n] = w[n]^l0  (16 x 64)
    #pragma unroll
    for (int s = 0; s < 2; ++s) {
        const int n = lane + 32 * s;
        float qr = 1.0f, qi = 0.0f;
        for (int l0 = 0; l0 < 16; ++l0) {
            sQr[l0 * 65 + n] = qr;
            sQi[l0 * 65 + n] = qi;
            cmul(qr, qi, wr[s], wi[s]);
        }
    }

    // Two passes over l1 (64 columns each) to bound LDS at ~41 KB
    for (int pass = 0; pass < 2; ++pass) {
        __syncthreads();
        // Build P[n, l1] = 2*C2[n] * W16[n]^l1 ; store (Pr, -Pi)
        #pragma unroll
        for (int s = 0; s < 2; ++s) {
            const int n = lane + 32 * s;
            for (int l1 = 0; l1 < 64; ++l1) {
                sPr[n * 65 + l1] = pr[s];
                sPi[n * 65 + l1] = -pi[s];
                cmul(pr[s], pi[s], W16r[s], W16i[s]);   // state carries across passes
            }
        }
        __syncthreads();

        // 4 output column tiles per pass; K-loop over N=64 in steps of 4
        for (int j = 0; j < 4; ++j) {
            v8f acc = {};
            #pragma unroll
            for (int k0 = 0; k0 < 64; k0 += 4) {
                const int kk = k0 + 2 * halfSel;   // A/B K-pair per lane half
                v2f a, b;
                // real part: Qr (A: 16x4) x Pr (B: 4x16)
                a.x = sQr[mn * 65 + kk];
                a.y = sQr[mn * 65 + kk + 1];
                b.x = sPr[kk * 65 + j * 16 + mn];
                b.y = sPr[(kk + 1) * 65 + j * 16 + mn];
                acc = __builtin_amdgcn_wmma_f32_16x16x4_f32(
                    false, a, false, b, (short)0, acc, false, false);
                // imag part: Qi x (-Pi), accumulated into same tile
                a.x = sQi[mn * 65 + kk];
                a.y = sQi[mn * 65 + kk + 1];
                b.x = sPi[kk * 65 + j * 16 + mn];
                b.y = sPi[(kk + 1) * 65 + j * 16 + mn];
                acc = __builtin_amdgcn_wmma_f32_16x16x4_f32(
                    false, a, false, b, (short)0, acc, false, false);
            }
            // D layout: VGPR r <-> M = r + 8*halfSel ; lane&15 <-> column l1
            // l = 16*l1 + M  -> each lane owns 8 contiguous floats: two b128 stores
            const int jg = pass * 4 + j;
            float* dst = out + h * L_DIM + (jg * 16 + mn) * 16 + 8 * halfSel;
            *reinterpret_cast<float4*>(dst)     = make_float4(acc[0], acc[1], acc[2], acc[3]);
            *reinterpret_cast<float4*>(dst + 4) = make_float4(acc[4], acc[5], acc[6], acc[7]);
        }
    }
}

extern "C" void kernel_launch(void* const* d_in, const int* in_sizes, int n_in,
                              void* d_out, int out_size, void* d_ws, size_t ws_size,
                              hipStream_t stream) {
    const float* log_dt     = (const float*)d_in[0];
    const float* C_real     = (const float*)d_in[1];
    const float* log_A_real = (const float*)d_in[2];
    const float* A_imag     = (const float*)d_in[3];
    // d_in[4] is L (=2048), fixed by the reference; baked in as L_DIM.
    float* out = (float*)d_out;

    dim3 grid(H_DIM);
    dim3 block(32);   // one wave32 per head; EXEC all-1s for WMMA
    hipLaunchKernelGGL(s4d_wmma_kernel, grid, block, 0, stream,
                       log_dt, C_real, log_A_real, A_imag, out);
}